// GCN_3650722201611
// MI455X (gfx1250) — compile-verified
//
#include <hip/hip_runtime.h>
#include <hip/hip_bf16.h>
#include <stdint.h>

typedef __attribute__((ext_vector_type(2))) float v2f;
typedef __attribute__((ext_vector_type(8))) float v8f;

// ---------------------------------------------------------------- utilities
__global__ void fill_zero_kernel(float* __restrict__ p, int n) {
    int i = blockIdx.x * blockDim.x + threadIdx.x;
    if (i < n) p[i] = 0.0f;
}

// deg[col[e]] += 1   (col = edge_index[1])
__global__ void degree_kernel(const long long* __restrict__ ei, float* __restrict__ deg, int E) {
    int e = blockIdx.x * blockDim.x + threadIdx.x;
    if (e < E) atomicAdd(&deg[(int)ei[(size_t)E + e]], 1.0f);
}

// deg -> (deg+1)^-1/2 in place
__global__ void dinv_kernel(float* __restrict__ deg, int N) {
    int i = blockIdx.x * blockDim.x + threadIdx.x;
    if (i < N) deg[i] = rsqrtf(deg[i] + 1.0f);
}

// ---------------------------------------------------------------- WMMA GEMM
// H[N,M] = X[N,K] @ W[K,M], fp32 via V_WMMA_F32_16X16X4_F32.
// One wave per 16x16 output tile. Block = (M/16) waves; blockIdx.x = row tile.
//
// fp32 A 16x4 fragment layout (ISA 7.12.2): lanes 0-15 M=lane {K=0,1 in v0,v1},
// lanes 16-31 M=lane-16 {K=2,3}.  B 4x16 mirrors (K along vgpr/half, N along lane).
// C/D: vgpr r, lanes 0-15 -> M=r, N=lane; lanes 16-31 -> M=r+8.
template <int K, int M>
__global__ void gemm_wmma_f32(const float* __restrict__ X, const float* __restrict__ W,
                              float* __restrict__ H, int N) {
    const int wave = threadIdx.x >> 5;          // column tile within row tile
    const int lane = threadIdx.x & 31;
    const int r    = lane & 15;
    const int half = lane >> 4;                 // 0 or 1
    const int rowBase = blockIdx.x * 16;
    const int colBase = wave * 16;

    int arow = rowBase + r;
    if (arow >= N) arow = N - 1;                // clamp keeps EXEC all-1s for WMMA
    const float* __restrict__ xrow = X + (size_t)arow * K;

    v8f acc = {};
#pragma unroll
    for (int k = 0; k < K; k += 4) {
        const int kk = k + 2 * half;
        v2f a, b;
        a.x = xrow[kk];
        a.y = xrow[kk + 1];
        b.x = W[(size_t)kk * M + colBase + r];
        b.y = W[(size_t)(kk + 1) * M + colBase + r];
        acc = __builtin_amdgcn_wmma_f32_16x16x4_f32(false, a, false, b,
                                                    (short)0, acc, false, false);
    }

#pragma unroll
    for (int vr = 0; vr < 8; ++vr) {
        const int row = rowBase + vr + 8 * half;
        if (row < N) H[(size_t)row * M + colBase + r] = acc[vr];
    }
}

// ---------------------------------------------------------------- edge aggregation
// AGG[col] += dinv[row]*dinv[col] * H[row]; (F/4) lanes per edge, float4 per lane.
__global__ void edge_agg_kernel(const float* __restrict__ H, const long long* __restrict__ ei,
                                const float* __restrict__ dinv, float* __restrict__ AGG,
                                int E, int lpeShift /* log2(F/4) */) {
    const int t = blockIdx.x * blockDim.x + threadIdx.x;
    const int e = t >> lpeShift;
    if (e >= E) return;
    const int f = (t & ((1 << lpeShift) - 1)) * 4;
    const int F = 4 << lpeShift;

    const int row = (int)ei[e];
    const int col = (int)ei[(size_t)E + e];
    const float norm = dinv[row] * dinv[col];

    const float4 hv = *(const float4*)(H + (size_t)row * F + f);
    float* dst = AGG + (size_t)col * F + f;
    atomicAdd(dst + 0, hv.x * norm);
    atomicAdd(dst + 1, hv.y * norm);
    atomicAdd(dst + 2, hv.z * norm);
    atomicAdd(dst + 3, hv.w * norm);
}

// AGG[i] = (AGG[i] + dinv[i]^2 * H[i] + b)   (+ ReLU)
__global__ void self_bias_act_kernel(const float* __restrict__ H, float* __restrict__ AGG,
                                     const float* __restrict__ bias, const float* __restrict__ dinv,
                                     int N, int fShift /* log2(F) */, int relu) {
    const int t = blockIdx.x * blockDim.x + threadIdx.x;
    const int i = t >> fShift;
    if (i >= N) return;
    const int f = t & ((1 << fShift) - 1);
    const size_t idx = ((size_t)i << fShift) + f;
    const float d = dinv[i];
    float v = AGG[idx] + d * d * H[idx] + bias[f];
    if (relu) v = fmaxf(v, 0.0f);
    AGG[idx] = v;
}

// ---------------------------------------------------------------- pooling + head
__global__ void pool_sum_kernel(const float* __restrict__ H, const long long* __restrict__ batch,
                                float* __restrict__ sums, int N) {
    const int t = blockIdx.x * blockDim.x + threadIdx.x;
    const int i = t >> 6;
    if (i >= N) return;
    const int f = t & 63;
    atomicAdd(&sums[(((int)batch[i]) << 6) + f], H[((size_t)i << 6) + f]);
}

__global__ void pool_cnt_kernel(const long long* __restrict__ batch, float* __restrict__ cnts, int N) {
    const int i = blockIdx.x * blockDim.x + threadIdx.x;
    if (i < N) atomicAdd(&cnts[(int)batch[i]], 1.0f);
}

// out[g,j] = dot(sums[g,:]/max(cnt,1), Wl[:,j]) + bl[j]   (G=512, 64->10)
__global__ void head_linear_kernel(const float* __restrict__ sums, const float* __restrict__ cnts,
                                   const float* __restrict__ Wl, const float* __restrict__ bl,
                                   float* __restrict__ out, int G) {
    const int t = blockIdx.x * blockDim.x + threadIdx.x;
    const int g = t / 10;
    if (g >= G) return;
    const int j = t - g * 10;
    const float inv = 1.0f / fmaxf(cnts[g], 1.0f);
    float acc = 0.0f;
#pragma unroll
    for (int k = 0; k < 64; ++k) acc += sums[(size_t)g * 64 + k] * Wl[k * 10 + j];
    out[(size_t)g * 10 + j] = acc * inv + bl[j];
}

// ---------------------------------------------------------------- launcher
static inline int cdiv(long long a, int b) { return (int)((a + b - 1) / b); }

extern "C" void kernel_launch(void* const* d_in, const int* in_sizes, int n_in,
                              void* d_out, int out_size, void* d_ws, size_t ws_size,
                              hipStream_t stream) {
    const float*     x   = (const float*)d_in[0];
    const long long* ei  = (const long long*)d_in[1];
    const long long* bat = (const long long*)d_in[2];
    const float* W1 = (const float*)d_in[3];  const float* b1 = (const float*)d_in[4];
    const float* W2 = (const float*)d_in[5];  const float* b2 = (const float*)d_in[6];
    const float* W3 = (const float*)d_in[7];  const float* b3 = (const float*)d_in[8];
    const float* Wl = (const float*)d_in[9];  const float* bl = (const float*)d_in[10];
    float* out = (float*)d_out;

    const int N = in_sizes[0] / 64;      // 50000
    const int E = in_sizes[1] / 2;       // 800000
    const int G = out_size / 10;         // 512

    // workspace layout (floats)
    float* ws   = (float*)d_ws;
    float* dinv = ws;                                   // N
    float* tmpA = ws + ((N + 63) & ~63);                // N*128  (X@W)
    float* aggB = tmpA + (size_t)N * 128;               // N*128  (aggregated)
    float* sums = aggB + (size_t)N * 128;               // G*64
    float* cnts = sums + (size_t)G * 64;                // G

    const int T = 256;

    // --- degrees ---
    fill_zero_kernel<<<cdiv(N, T), T, 0, stream>>>(dinv, N);
    degree_kernel<<<cdiv(E, T), T, 0, stream>>>(ei, dinv, E);
    dinv_kernel<<<cdiv(N, T), T, 0, stream>>>(dinv, N);

    const int rowTiles = cdiv(N, 16);

    // --- layer 1: 64 -> 128, ReLU ---
    gemm_wmma_f32<64, 128><<<rowTiles, 8 * 32, 0, stream>>>(x, W1, tmpA, N);
    fill_zero_kernel<<<cdiv((long long)N * 128, T), T, 0, stream>>>(aggB, N * 128);
    edge_agg_kernel<<<cdiv((long long)E * 32, T), T, 0, stream>>>(tmpA, ei, dinv, aggB, E, 5);
    self_bias_act_kernel<<<cdiv((long long)N * 128, T), T, 0, stream>>>(tmpA, aggB, b1, dinv, N, 7, 1);

    // --- layer 2: 128 -> 128, ReLU ---
    gemm_wmma_f32<128, 128><<<rowTiles, 8 * 32, 0, stream>>>(aggB, W2, tmpA, N);
    fill_zero_kernel<<<cdiv((long long)N * 128, T), T, 0, stream>>>(aggB, N * 128);
    edge_agg_kernel<<<cdiv((long long)E * 32, T), T, 0, stream>>>(tmpA, ei, dinv, aggB, E, 5);
    self_bias_act_kernel<<<cdiv((long long)N * 128, T), T, 0, stream>>>(tmpA, aggB, b2, dinv, N, 7, 1);

    // --- layer 3: 128 -> 64, no activation ---
    gemm_wmma_f32<128, 64><<<rowTiles, 4 * 32, 0, stream>>>(aggB, W3, tmpA, N);
    fill_zero_kernel<<<cdiv((long long)N * 64, T), T, 0, stream>>>(aggB, N * 64);
    edge_agg_kernel<<<cdiv((long long)E * 16, T), T, 0, stream>>>(tmpA, ei, dinv, aggB, E, 4);
    self_bias_act_kernel<<<cdiv((long long)N * 64, T), T, 0, stream>>>(tmpA, aggB, b3, dinv, N, 6, 0);

    // --- global mean pool + linear head ---
    fill_zero_kernel<<<cdiv(G * 64 + G, T), T, 0, stream>>>(sums, G * 64 + G);
    pool_sum_kernel<<<cdiv((long long)N * 64, T), T, 0, stream>>>(aggB, bat, sums, N);
    pool_cnt_kernel<<<cdiv(N, T), T, 0, stream>>>(bat, cnts, N);
    head_linear_kernel<<<cdiv((long long)G * 10, T), T, 0, stream>>>(sums, cnts, Wl, bl, out, G);
}